// SpikesToTimesDecoder_42777874268317
// MI455X (gfx1250) — compile-verified
//
#include <hip/hip_runtime.h>

// ---- problem constants (from reference) ----
#define B_DIM 64
#define T_DIM 2048
#define O_DIM 256
#define K_SPK 16
#define DT_F  0.001f

// ---- tiling ----
#define CH 128                  // channels per block (block covers half of O)
#define CT 32                   // time steps staged per chunk (== mask width)
#define NCHUNK (T_DIM / CT)     // 64 chunks
#define ISSUES (CT / 4)         // 8 async b128 issues per thread per chunk

#ifdef __has_builtin
# if __has_builtin(__builtin_amdgcn_global_load_async_to_lds_b128) && \
     __has_builtin(__builtin_amdgcn_s_wait_asynccnt)
#  define USE_ASYNC_LDS 1
# endif
#endif

typedef int v4i __attribute__((vector_size(16)));

__global__ __launch_bounds__(CH) void spikes_to_times_kernel(
    const float* __restrict__ spikes, float* __restrict__ out)
{
    __shared__ float sbuf[2][CT * CH];   // 2 x 16 KB double buffer
    __shared__ int   s_notdone;

    const int tid = threadIdx.x;
    const int b   = blockIdx.x >> 1;             // batch index
    const int ob  = (blockIdx.x & 1) * CH;       // channel base (0 or 128)
    const float* gbase = spikes + (size_t)b * T_DIM * O_DIM + ob;

    // staging geometry: wave (32 lanes) x 16B = one 512B row (128 ch * 4B)
    const int r0  = tid >> 5;        // row-within-group handled by this wave
    const int col = (tid & 31) << 2; // float column of this lane's 16B piece

    // earliest-K spike times, kept entirely in registers
    float tk[K_SPK];
#pragma unroll
    for (int k = 0; k < K_SPK; ++k) tk[k] = __builtin_inff();
    int cnt = 0;

    // sparse extraction of set bits: runs at most K_SPK times over the
    // whole kernel, so the 16-way register select is off the hot path
    auto extract = [&](unsigned mask, int t0) {
        while (mask != 0u && cnt < K_SPK) {
            const int t = __builtin_ctz(mask);
            mask &= mask - 1u;
#pragma unroll
            for (int k = 0; k < K_SPK; ++k)
                if (cnt == k) tk[k] = (float)(t0 + t);
            ++cnt;
        }
    };

#if USE_ASYNC_LDS
    auto stage = [&](int c, int bi) {
#pragma unroll
        for (int j = 0; j < ISSUES; ++j) {
            const int r = j * 4 + r0;  // 4 waves cover 4 rows per issue round
            __builtin_amdgcn_global_load_async_to_lds_b128(
                (v4i*)(gbase + (size_t)(c * CT + r) * O_DIM + col),
                (v4i*)(&sbuf[bi][r * CH + col]),
                0 /*offset*/, 0 /*cpol*/);
        }
    };

    stage(0, 0);
    for (int c = 0; c < NCHUNK; ++c) {
        const int cur = c & 1;
        if (c + 1 < NCHUNK) {
            stage(c + 1, cur ^ 1);                    // prefetch next chunk
            __builtin_amdgcn_s_wait_asynccnt(ISSUES); // chunk c done, next in flight
        } else {
            __builtin_amdgcn_s_wait_asynccnt(0);
        }
        if (tid == 0) s_notdone = 0;
        __syncthreads();  // all waves' async LDS writes for chunk c visible

        // build 32-bit spike mask for this thread's channel (~3 VALU/elem)
        unsigned mask = 0u;
#pragma unroll
        for (int t = 0; t < CT; ++t)
            mask |= (sbuf[cur][t * CH + tid] != 0.0f) ? (1u << t) : 0u;

        extract(mask, c * CT);

        if (cnt < K_SPK) s_notdone = 1;  // benign same-value race
        __syncthreads();                 // also fences reads before next overwrite
        if (s_notdone == 0) break;       // uniform block-wide early exit
    }
#else
    // Fallback: synchronous staged copy (no async builtins on this toolchain)
    for (int c = 0; c < NCHUNK; ++c) {
        if (tid == 0) s_notdone = 0;
#pragma unroll
        for (int j = 0; j < ISSUES; ++j) {
            const int r = j * 4 + r0;
            const float4 v = *(const float4*)(gbase + (size_t)(c * CT + r) * O_DIM + col);
            *(float4*)(&sbuf[0][r * CH + col]) = v;
        }
        __syncthreads();
        unsigned mask = 0u;
#pragma unroll
        for (int t = 0; t < CT; ++t)
            mask |= (sbuf[0][t * CH + tid] != 0.0f) ? (1u << t) : 0u;
        extract(mask, c * CT);
        if (cnt < K_SPK) s_notdone = 1;
        __syncthreads();
        if (s_notdone == 0) break;
    }
#endif

    // out[b, k, o] = k-th earliest spike time * DT  (inf if missing)
    float* op = out + (size_t)b * K_SPK * O_DIM + ob + tid;
#pragma unroll
    for (int k = 0; k < K_SPK; ++k)
        op[(size_t)k * O_DIM] = tk[k] * DT_F;
}

extern "C" void kernel_launch(void* const* d_in, const int* in_sizes, int n_in,
                              void* d_out, int out_size, void* d_ws, size_t ws_size,
                              hipStream_t stream) {
    (void)in_sizes; (void)n_in; (void)d_ws; (void)ws_size; (void)out_size;
    const float* spikes = (const float*)d_in[0];
    float* out = (float*)d_out;
    dim3 grid(B_DIM * (O_DIM / CH));   // 128 blocks
    dim3 block(CH);                    // 128 threads = 4 wave32
    spikes_to_times_kernel<<<grid, block, 0, stream>>>(spikes, out);
}